// Attention_86423331930617
// MI455X (gfx1250) — compile-verified
//
#include <hip/hip_runtime.h>

#define BATCH 16
#define SEQ   600
#define DIM   128
#define HEADS 8
#define DHEAD 32
#define INNER 256
#define D3    384
#define NC1   98304    // 3*DIM*INNER
#define NC2   32768    // INNER*DIM
#define MT    38       // ceil(600/16)
#define SPAD  608      // 38*16

typedef float v2f __attribute__((ext_vector_type(2)));
typedef float v8f __attribute__((ext_vector_type(8)));
typedef unsigned int v4u __attribute__((ext_vector_type(4)));
typedef int v4i __attribute__((ext_vector_type(4)));
typedef int v8i __attribute__((ext_vector_type(8)));

static __device__ __forceinline__ v8f wmma_f32(v2f a, v2f b, v8f c) {
  // V_WMMA_F32_16X16X4_F32: D = A(16x4) * B(4x16) + C(16x16), fp32 throughout
  return __builtin_amdgcn_wmma_f32_16x16x4_f32(false, a, false, b, (short)0, c,
                                               false, false);
}

// ---------------------------------------------------------------------------
// Tensor Data Mover: 1-D copy global -> LDS (n_elems f32, contiguous).
// D# packing per CDNA5 ISA ch.8: group0 {count=1, lds_addr, global_addr, type=2},
// group1 {data_size=4B, tensor_dim0=n, tensor_dim1=1, tile_dim0=n, stride0=n}.
// ---------------------------------------------------------------------------
static __device__ __forceinline__ void tdm_load_1d(const float* gsrc, void* ldst,
                                                   int n_elems) {
  const unsigned long long ga = (unsigned long long)(__SIZE_TYPE__)gsrc;
  const unsigned lds = (unsigned)(__SIZE_TYPE__)ldst;
  const unsigned n = (unsigned)n_elems;
  v4u g0;
  g0[0] = 1u;                                              // count=1, user mode
  g0[1] = lds;                                             // lds_addr
  g0[2] = (unsigned)(ga & 0xffffffffu);                    // global_addr[31:0]
  g0[3] = (unsigned)((ga >> 32) & 0x01ffffffu) | (2u << 30); // addr[56:32]|type=2
  v8i g1;
  g1[0] = (int)(2u << 16);                                 // data_size = 4 bytes
  g1[1] = (int)((n & 0xffffu) << 16);                      // tensor_dim0[15:0]
  g1[2] = (int)(((n >> 16) & 0xffffu) | (1u << 16));       // dim0[31:16]|dim1=1
  g1[3] = (int)((n & 0xffffu) << 16);                      // tile_dim0 = n
  g1[4] = 0;                                               // tile_dim1/2 unused
  g1[5] = (int)n;                                          // dim0_stride[31:0]
  g1[6] = 0;
  g1[7] = 0;
  const v4i gz = {0, 0, 0, 0};
#if defined(__clang_major__) && (__clang_major__ >= 23)
  const v8i gz8 = {0, 0, 0, 0, 0, 0, 0, 0};
  __builtin_amdgcn_tensor_load_to_lds(g0, g1, gz, gz, gz8, 0);
#else
  __builtin_amdgcn_tensor_load_to_lds(g0, g1, gz, gz, 0);
#endif
}

// ---------------------------------------------------------------------------
// Kernel A: tiny hypernet chains; outputs stored TRANSPOSED [k][b] so the big
// skinny GEMM can TDM-copy them straight into LDS.
// ---------------------------------------------------------------------------
__global__ void k_hyper_small(const float* __restrict__ fr,
                              const float* __restrict__ rs,
                              const float* __restrict__ W1, const float* __restrict__ b1,
                              const float* __restrict__ W2, const float* __restrict__ b2,
                              const float* __restrict__ V1, const float* __restrict__ c1,
                              const float* __restrict__ V2, const float* __restrict__ c2,
                              float* __restrict__ t2T, float* __restrict__ u2T) {
  __shared__ float s_t1[BATCH * D3];
  __shared__ float s_u1[BATCH * INNER];
  const int j = threadIdx.x;
  if (j < D3) {
    const float w0 = W1[j], w1 = W1[D3 + j], bb = b1[j];
#pragma unroll
    for (int b = 0; b < BATCH; ++b)
      s_t1[b * D3 + j] = fr[b] * w0 + rs[b] * w1 + bb;
  }
  if (j < INNER) {
    const float w0 = V1[j], w1 = V1[INNER + j], bb = c1[j];
#pragma unroll
    for (int b = 0; b < BATCH; ++b)
      s_u1[b * INNER + j] = fr[b] * w0 + rs[b] * w1 + bb;
  }
  __syncthreads();
  if (j < D3) {
    float acc[BATCH];
#pragma unroll
    for (int b = 0; b < BATCH; ++b) acc[b] = 0.f;
    for (int k = 0; k < D3; ++k) {
      const float w = W2[k * D3 + j];
#pragma unroll
      for (int b = 0; b < BATCH; ++b) acc[b] += s_t1[b * D3 + k] * w;
    }
    const float bb = b2[j];
#pragma unroll
    for (int b = 0; b < BATCH; ++b) t2T[j * BATCH + b] = acc[b] + bb;
  }
  if (j < INNER) {
    float acc[BATCH];
#pragma unroll
    for (int b = 0; b < BATCH; ++b) acc[b] = 0.f;
    for (int k = 0; k < INNER; ++k) {
      const float w = V2[k * INNER + j];
#pragma unroll
      for (int b = 0; b < BATCH; ++b) acc[b] += s_u1[b * INNER + k] * w;
    }
    const float bb = c2[j];
#pragma unroll
    for (int b = 0; b < BATCH; ++b) u2T[j * BATCH + b] = acc[b] + bb;
  }
}

// ---------------------------------------------------------------------------
// Kernel B: skinny GEMM  out[b, j] = dot(t2T[:, b], W[:, j]) + bias[j]
// W (row-major Kd x NC, 151/33 MB) streamed once, coalesced + prefetch.
// t2T [k][b] staged into LDS via the Tensor Data Mover (async, TENSORcnt).
// ---------------------------------------------------------------------------
__global__ void k_hyper_big(const float* __restrict__ t2T, int Kd,
                            const float* __restrict__ W,
                            const float* __restrict__ bias,
                            float* __restrict__ out, int NC) {
  extern __shared__ float st[];  // [k][b], Kd*16 floats
  if (threadIdx.x == 0) {
    tdm_load_1d(t2T, st, Kd * BATCH);
    __builtin_amdgcn_s_wait_tensorcnt(0);
  }
  __syncthreads();
  const size_t j = (size_t)blockIdx.x * blockDim.x + threadIdx.x;
  float acc[BATCH];
#pragma unroll
  for (int b = 0; b < BATCH; ++b) acc[b] = 0.f;
  const float* wp = W + j;
  for (int k = 0; k < Kd; ++k) {
    if ((k & 7) == 0 && k + 16 < Kd)
      __builtin_prefetch(wp + (size_t)16 * NC, 0, 1);
    const float w = *wp;
    wp += NC;
#pragma unroll
    for (int b = 0; b < BATCH; ++b) acc[b] += st[k * BATCH + b] * w;
  }
  const float bb = bias[j];
#pragma unroll
  for (int b = 0; b < BATCH; ++b) out[(size_t)b * NC + j] = acc[b] + bb;
}

// ---------------------------------------------------------------------------
// Kernel C: strided softmax, writing a TRANSPOSED (K-contiguous) copy:
//   outT[b*perB + c*len + i] = softmax_i( in[b*perB + c + i*strideIn] )
// ---------------------------------------------------------------------------
__global__ void k_softmax_T(const float* __restrict__ in, float* __restrict__ outT,
                            unsigned long long perB, int ncolsPerB,
                            int strideIn, int len, int total) {
  const int idx = blockIdx.x * blockDim.x + threadIdx.x;
  if (idx >= total) return;
  const int b = idx / ncolsPerB, c = idx % ncolsPerB;
  const float* p = in + (size_t)b * perB + c;
  float mx = -3.4e38f;
  for (int i = 0; i < len; ++i) mx = fmaxf(mx, p[(size_t)i * strideIn]);
  float s = 0.f;
  for (int i = 0; i < len; ++i) s += __expf(p[(size_t)i * strideIn] - mx);
  const float inv = 1.f / s;
  float* q = outT + (size_t)b * perB + (size_t)c * len;
  for (int i = 0; i < len; ++i) q[i] = __expf(p[(size_t)i * strideIn] - mx) * inv;
}

// ---------------------------------------------------------------------------
// Kernel D: batched f32 WMMA GEMM  C[b] = A[b](M x Kd) * BT[b](N x Kd)^T
// Both operands K-contiguous -> 2x global_load_b64 per WMMA, pointer-increment
// only (no per-iteration address mul, no EXEC toggling: OOB rows clamped).
// ---------------------------------------------------------------------------
__global__ void k_gemm_wmma(const float* __restrict__ A, unsigned long long sA, int lda,
                            const float* __restrict__ BT, unsigned long long sB, int ldbt,
                            float* __restrict__ C, unsigned long long sC, int ldc,
                            int M, int Kd) {
  const int b    = blockIdx.z;
  const int wid  = threadIdx.x >> 5;
  const int lane = threadIdx.x & 31;
  const int half = lane >> 4;
  const int lid  = lane & 15;
  const int nt   = blockIdx.y * 8 + wid;
  const int m0   = blockIdx.x * 16, n0 = nt * 16;
  const int mrow = m0 + lid, ncol = n0 + lid;
  const int mcl  = (mrow < M) ? mrow : (M - 1);   // clamp: pad rows never stored
  const int koff = half * 2;
  const float* ap = A + (size_t)b * sA + (size_t)mcl * lda + koff;
  const float* bp = BT + (size_t)b * sB + (size_t)ncol * ldbt + koff;
  v8f acc = {0.f, 0.f, 0.f, 0.f, 0.f, 0.f, 0.f, 0.f};
#pragma unroll 8
  for (int kc = 0; kc < Kd; kc += 4) {
    const v2f a  = *(const v2f*)ap;
    const v2f bv = *(const v2f*)bp;
    acc = wmma_f32(a, bv, acc);
    ap += 4;
    bp += 4;
  }
  float* Cp = C + (size_t)b * sC;
#pragma unroll
  for (int i = 0; i < 8; ++i) {
    const int r = m0 + i + 8 * half;
    if (r < M) Cp[(size_t)r * ldc + ncol] = acc[i];
  }
}

// ---------------------------------------------------------------------------
// Kernel E: attention, one (b,h) per block; 128 threads = 4 waves; each wave
// owns a 16x608 f32 score strip in dynamic LDS (152 KB / workgroup).
// S = scale*Q*K^T (WMMA) -> row softmax (32 lanes, shfl_xor combine) * mask
// -> O = P*V (WMMA, P read from LDS as contiguous b64).
// ---------------------------------------------------------------------------
__global__ void k_attention(const float* __restrict__ qkv,   // [3][B][SEQ][INNER]
                            const float* __restrict__ mask,  // [B][SEQ][SEQ]
                            float* __restrict__ aout) {      // [B][SEQ][INNER]
  extern __shared__ float smem[];
  const int b = blockIdx.x >> 3, h = blockIdx.x & 7;
  const int wid  = threadIdx.x >> 5;
  const int lane = threadIdx.x & 31;
  const int half = lane >> 4;
  const int lid  = lane & 15;
  const int koff = half * 2;
  const size_t QKVSZ = (size_t)BATCH * SEQ * INNER;
  const float* Qp = qkv + 0 * QKVSZ + (size_t)b * SEQ * INNER + h * DHEAD;
  const float* Kp = qkv + 1 * QKVSZ + (size_t)b * SEQ * INNER + h * DHEAD;
  const float* Vp = qkv + 2 * QKVSZ + (size_t)b * SEQ * INNER + h * DHEAD;
  const float* maskp = mask + (size_t)b * SEQ * SEQ;
  float* sc = smem + (size_t)wid * 16 * SPAD;
  const float scale = 0.17677669529663687f;  // 1/sqrt(32)

  for (int it = 0; it < 10; ++it) {          // ceil(38 m-tiles / 4 waves)
    const int mt = it * 4 + wid;
    const bool act = (mt < MT);
    const int m0 = mt * 16;
    const int mrow = m0 + lid;
    const int mcl = (mrow < SEQ) ? mrow : (SEQ - 1);
    if (act) {
      const float* qp = Qp + (size_t)mcl * INNER + koff;  // K-contiguous
      for (int nt = 0; nt < MT; ++nt) {                   // S: 38 n-tiles
        const int col = nt * 16 + lid;
        const int ccl = (col < SEQ) ? col : (SEQ - 1);    // pad cols never read
        const float* kp = Kp + (size_t)ccl * INNER + koff;
        v8f acc = {0.f, 0.f, 0.f, 0.f, 0.f, 0.f, 0.f, 0.f};
#pragma unroll
        for (int kc = 0; kc < DHEAD; kc += 4)
          acc = wmma_f32(*(const v2f*)(qp + kc), *(const v2f*)(kp + kc), acc);
#pragma unroll
        for (int i = 0; i < 8; ++i)
          sc[(i + 8 * half) * SPAD + col] = acc[i] * scale;
      }
    }
    __syncthreads();
    if (act) {                               // softmax * mask, all 32 lanes
      float* row = sc + lid * SPAD;
      const int j0 = half * 300, j1 = j0 + 300;
      float mp = -3.4e38f;
      for (int jj = j0; jj < j1; ++jj) mp = fmaxf(mp, row[jj]);
      const float mx = fmaxf(mp, __shfl_xor(mp, 16, 32));
      float sp = 0.f;
      for (int jj = j0; jj < j1; ++jj) sp += __expf(row[jj] - mx);
      const float s = sp + __shfl_xor(sp, 16, 32);
      const float inv = 1.f / s;
      if (mrow < SEQ) {
        const float* mrp = maskp + (size_t)mrow * SEQ;
        for (int jj = j0; jj < j1; ++jj)
          row[jj] = __expf(row[jj] - mx) * inv * mrp[jj];
      } else {
        for (int jj = j0; jj < j1; ++jj) row[jj] = 0.f;
      }
    }
    __syncthreads();
    if (act) {                               // O = P @ V over K = [0,600)
      for (int nt2 = 0; nt2 < 2; ++nt2) {
        const int col2 = nt2 * 16 + lid;
        const float* pl = sc + lid * SPAD + koff;      // P row, LDS b64 reads
        const float* vp = Vp + (size_t)koff * INNER + col2;
        v8f acc = {0.f, 0.f, 0.f, 0.f, 0.f, 0.f, 0.f, 0.f};
#pragma unroll 5
        for (int kc = 0; kc < SEQ; kc += 4) {          // 600 % 4 == 0, no pad
          v2f bv;
          bv[0] = vp[0];
          bv[1] = vp[INNER];
          acc = wmma_f32(*(const v2f*)pl, bv, acc);
          pl += 4;
          vp += 4 * INNER;
        }
#pragma unroll
        for (int i = 0; i < 8; ++i) {
          const int r = m0 + i + 8 * half;
          if (r < SEQ)
            aout[((size_t)b * SEQ + r) * INNER + h * DHEAD + col2] = acc[i];
        }
      }
    }
    __syncthreads();
  }
}

// ---------------------------------------------------------------------------
extern "C" void kernel_launch(void* const* d_in, const int* in_sizes, int n_in,
                              void* d_out, int out_size, void* d_ws, size_t ws_size,
                              hipStream_t stream) {
  (void)in_sizes; (void)n_in; (void)out_size; (void)ws_size;
  const float* x    = (const float*)d_in[0];
  const float* mask = (const float*)d_in[1];
  const float* rs   = (const float*)d_in[2];
  const float* fr   = (const float*)d_in[3];
  const float* W1 = (const float*)d_in[4];  const float* b1 = (const float*)d_in[5];
  const float* W2 = (const float*)d_in[6];  const float* b2 = (const float*)d_in[7];
  const float* W3 = (const float*)d_in[8];  const float* b3 = (const float*)d_in[9];
  const float* V1 = (const float*)d_in[10]; const float* c1 = (const float*)d_in[11];
  const float* V2 = (const float*)d_in[12]; const float* c2 = (const float*)d_in[13];
  const float* V3 = (const float*)d_in[14]; const float* c3 = (const float*)d_in[15];
  float* out = (float*)d_out;

  float* ws    = (float*)d_ws;
  float* t2T   = ws;                                    // 384*16 (transposed)
  float* u2T   = t2T + D3 * BATCH;                      // 256*16 (transposed)
  float* hraw  = u2T + INNER * BATCH;                   // 16*98304
  float* qkvwT = hraw + (size_t)BATCH * NC1;            // 16*768*128 (softmaxed, T)
  float* owraw = qkvwT + (size_t)BATCH * NC1;           // 16*32768
  float* owT   = owraw + (size_t)BATCH * NC2;           // 16*128*256 (softmaxed, T)
  float* qkv   = owT + (size_t)BATCH * NC2;             // 3*16*600*256
  float* aout  = qkv + (size_t)3 * BATCH * SEQ * INNER; // 16*600*256

  // 1) small hypernet chains (transposed outputs for TDM staging)
  k_hyper_small<<<1, D3, 0, stream>>>(fr, rs, W1, b1, W2, b2, V1, c1, V2, c2,
                                      t2T, u2T);

  // 2) h = t2 @ W3 + b3 (151 MB stream), then softmax over DIM axis ->
  //    transposed weights qkvwT[b][e][d] (K-contiguous for WMMA B operand)
  k_hyper_big<<<NC1 / 256, 256, D3 * BATCH * 4, stream>>>(t2T, D3, W3, b3, hraw, NC1);
  k_softmax_T<<<(BATCH * 3 * INNER + 255) / 256, 256, 0, stream>>>(
      hraw, qkvwT, NC1, 3 * INNER, 3 * INNER, DIM, BATCH * 3 * INNER);

  // 3) ow = u2 @ V3 + c3, softmax over INNER axis -> owT[b][d][e]
  k_hyper_big<<<NC2 / 256, 256, INNER * BATCH * 4, stream>>>(u2T, INNER, V3, c3,
                                                             owraw, NC2);
  k_softmax_T<<<(BATCH * DIM + 255) / 256, 256, 0, stream>>>(
      owraw, owT, NC2, DIM, DIM, INNER, BATCH * DIM);

  // 4) q/k/v projections: [600x128] @ [128x256] per batch (WMMA f32)
  const size_t QKVSZ = (size_t)BATCH * SEQ * INNER;
  for (int w = 0; w < 3; ++w) {
    k_gemm_wmma<<<dim3(MT, 2, BATCH), 256, 0, stream>>>(
        x, (unsigned long long)SEQ * DIM, DIM,
        qkvwT + (size_t)w * INNER * DIM, (unsigned long long)NC1, DIM,
        qkv + (size_t)w * QKVSZ, (unsigned long long)SEQ * INNER, INNER,
        SEQ, DIM);
  }

  // 5) attention per (b, h)
  k_attention<<<BATCH * HEADS, 128, 4 * 16 * SPAD * sizeof(float), stream>>>(
      qkv, mask, aout);

  // 6) final projection: [600x256] @ [256x128] per batch (WMMA f32)
  k_gemm_wmma<<<dim3(MT, 1, BATCH), 256, 0, stream>>>(
      aout, (unsigned long long)SEQ * INNER, INNER,
      owT, (unsigned long long)NC2, INNER,
      out, (unsigned long long)SEQ * DIM, DIM,
      SEQ, INNER);
}